// HybridScoring_40475771798061
// MI455X (gfx1250) — compile-verified
//
#include <hip/hip_runtime.h>
#include <math.h>

typedef __attribute__((ext_vector_type(2))) float v2f;
typedef __attribute__((ext_vector_type(8))) float v8f;

#define BATCH   16
#define NP1     10001
#define DIM     128
#define KNN     16
#define NEG_INF_F (-1e9f)
#define BPAD    132   // 128 + 4 pad floats -> LDS bank spread

// ---------------------------------------------------------------------------
// Kernel 1: per (batch, 16-node tile) scoring via WMMA diagonal trick.
//   score_raw[b,n] = dot(psi[b,n], query[b] + lam*nbsum[b,n])
//                    - mu*dist[b,n] + nu*demand[b,n]   (masked -> -1e9)
// One wave32 per tile.
// ---------------------------------------------------------------------------
__global__ __launch_bounds__(32) void hs_score_kernel(
    const float* __restrict__ query,      // [B, D]
    const float* __restrict__ psi,        // [B, NP1, D]
    const int*   __restrict__ knn,        // [B, NP1, K]
    const unsigned char* __restrict__ mask, // [B, NP1]
    const float* __restrict__ cur,        // [B, 2]
    const float* __restrict__ coords,     // [B, NP1, 2]
    const float* __restrict__ demands,    // [B, NP1]
    const float* __restrict__ lam_p,
    const float* __restrict__ mu_p,
    const float* __restrict__ nu_p,
    float* __restrict__ scores)           // [B, NP1] workspace
{
    __shared__ __align__(16) float bsum[16 * BPAD]; // B-matrix source: q + lam*nbsum
    __shared__ __align__(16) float Cs[16 * 16];     // staged C tile

    const int lane = threadIdx.x;           // 0..31
    const int tile = blockIdx.x;            // node tile
    const int b    = blockIdx.y;            // batch
    const int n0   = tile * 16;

    const float lam = fminf(fmaxf(lam_p[0], -2.0f), 3.0f);
    const float mu  = fminf(fmaxf(mu_p[0],   0.0f), 20.0f);
    const float nu  = fminf(fmaxf(nu_p[0],  -2.0f), 3.0f);

    const float* qb = query + b * DIM;
    const float* pb = psi + (size_t)b * NP1 * DIM;

    // ---- Phase A: build bsum[j][k] = q[k] + lam * sum_t psi[knn[n0+j][t]][k]
    // Each lane owns 4 consecutive k values.
    const float4 q4 = *(const float4*)(qb + lane * 4);
    for (int j = 0; j < 16; ++j) {
        int n = n0 + j; if (n >= NP1) n = NP1 - 1;           // uniform clamp
        const int* kj = knn + ((size_t)b * NP1 + n) * KNN;   // uniform -> s_load
        float4 acc = make_float4(0.f, 0.f, 0.f, 0.f);
        #pragma unroll
        for (int t = 0; t < KNN; ++t) {
            const int idx = kj[t];                            // wave-uniform
            const float4 nb = *(const float4*)(pb + (size_t)idx * DIM + lane * 4);
            acc.x += nb.x; acc.y += nb.y; acc.z += nb.z; acc.w += nb.w;
        }
        float4 bv;
        bv.x = q4.x + lam * acc.x;
        bv.y = q4.y + lam * acc.y;
        bv.z = q4.z + lam * acc.z;
        bv.w = q4.w + lam * acc.w;
        *(float4*)(&bsum[j * BPAD + lane * 4]) = bv;
    }
    __syncthreads();

    // ---- Phase B: 32 x V_WMMA_F32_16X16X4_F32 accumulation over K=128.
    // A (16x4 f32) layout: lanes 0-15 -> M=lane, K={c*4+0, c*4+1};
    //                      lanes 16-31 -> M=lane-16, K={c*4+2, c*4+3}.
    // B (4x16) symmetric: N = lane&15, same K halves.
    const int mrow  = lane & 15;
    const int khalf = (lane >> 4) * 2;      // 0 or 2
    int arow = n0 + mrow; if (arow >= NP1) arow = NP1 - 1;   // per-lane select
    const float* aptr = pb + (size_t)arow * DIM;
    const float* bptr = &bsum[mrow * BPAD];

    v8f acc = {};
    #pragma unroll 8
    for (int c = 0; c < 32; ++c) {
        const int k0 = c * 4 + khalf;
        const v2f a  = *(const v2f*)(aptr + k0);   // global_load_b64
        const v2f bb = *(const v2f*)(bptr + k0);   // ds_load_b64
        acc = __builtin_amdgcn_wmma_f32_16x16x4_f32(
            /*neg_a=*/false, a, /*neg_b=*/false, bb,
            /*c_mod=*/(short)0, acc, /*reuse_a=*/false, /*reuse_b=*/false);
    }

    // ---- Stage C tile to LDS, read diagonal, finish score.
    #pragma unroll
    for (int r = 0; r < 8; ++r) {
        const int row = (lane < 16) ? r : (r + 8);
        Cs[row * 16 + mrow] = acc[r];
    }
    __syncthreads();

    if (lane < 16) {
        const int n = n0 + lane;
        if (n < NP1) {
            float s = Cs[lane * 16 + lane];   // context + lam*interference
            const float dx = coords[((size_t)b * NP1 + n) * 2 + 0] - cur[b * 2 + 0];
            const float dy = coords[((size_t)b * NP1 + n) * 2 + 1] - cur[b * 2 + 1];
            s -= mu * sqrtf(dx * dx + dy * dy);
            s += nu * demands[(size_t)b * NP1 + n];   // capacity = 1.0
            if (mask[(size_t)b * NP1 + n]) s = NEG_INF_F;
            scores[(size_t)b * NP1 + n] = s;
        }
    }
}

// ---------------------------------------------------------------------------
// Kernel 2: row-wise log_softmax over NP1 elements. One block per batch.
// ---------------------------------------------------------------------------
__global__ __launch_bounds__(256) void hs_logsoftmax_kernel(
    const float* __restrict__ scores,  // [B, NP1]
    float* __restrict__ out)           // [B, NP1]
{
    __shared__ float red[256];
    const int b   = blockIdx.x;
    const int tid = threadIdx.x;
    const float* s = scores + (size_t)b * NP1;
    float* o = out + (size_t)b * NP1;

    // pass 1: max
    float m = -INFINITY;
    for (int i = tid; i < NP1; i += 256) m = fmaxf(m, s[i]);
    red[tid] = m;
    __syncthreads();
    #pragma unroll
    for (int off = 128; off > 0; off >>= 1) {
        if (tid < off) red[tid] = fmaxf(red[tid], red[tid + off]);
        __syncthreads();
    }
    m = red[0];
    __syncthreads();

    // pass 2: sum of exp
    float sum = 0.f;
    for (int i = tid; i < NP1; i += 256) sum += __expf(s[i] - m);
    red[tid] = sum;
    __syncthreads();
    #pragma unroll
    for (int off = 128; off > 0; off >>= 1) {
        if (tid < off) red[tid] += red[tid + off];
        __syncthreads();
    }
    const float lse = m + __logf(red[0]);

    for (int i = tid; i < NP1; i += 256) o[i] = s[i] - lse;
}

// ---------------------------------------------------------------------------
extern "C" void kernel_launch(void* const* d_in, const int* in_sizes, int n_in,
                              void* d_out, int out_size, void* d_ws, size_t ws_size,
                              hipStream_t stream) {
    const float*         query   = (const float*)d_in[0];
    const float*         psi     = (const float*)d_in[1];
    const int*           knn     = (const int*)d_in[2];
    const unsigned char* mask    = (const unsigned char*)d_in[3];
    const float*         cur     = (const float*)d_in[4];
    const float*         coords  = (const float*)d_in[5];
    const float*         demands = (const float*)d_in[6];
    const float*         lam     = (const float*)d_in[7];
    const float*         mu      = (const float*)d_in[8];
    const float*         nu      = (const float*)d_in[9];

    float* scores = (float*)d_ws;   // [BATCH, NP1] raw scores

    dim3 grid1((NP1 + 15) / 16, BATCH);
    hs_score_kernel<<<grid1, 32, 0, stream>>>(query, psi, knn, mask, cur, coords,
                                              demands, lam, mu, nu, scores);

    hs_logsoftmax_kernel<<<BATCH, 256, 0, stream>>>(scores, (float*)d_out);
}